// ModelNew_19688130085509
// MI455X (gfx1250) — compile-verified
//
#include <hip/hip_runtime.h>
#include <hip/hip_bf16.h>

typedef __attribute__((ext_vector_type(2))) float v2f;
typedef __attribute__((ext_vector_type(8))) float v8f;

#define CIN    3
#define COUT   16
#define IN_D   48
#define OUT_D  47
#define NCHUNK 6           // K = 24 = 6 chunks of 4
#define OUT_CH_STRIDE (OUT_D*OUT_D*OUT_D)   // 103823
#define ROWS   (8*OUT_D*OUT_D)              // 17672

// ---------------------------------------------------------------------------
// Prep: fold ConvTranspose3d + scale1 + avgpool(0.125) + scale2 into an
// effective 2x2x2 conv kernel, laid out ready to load as WMMA A-fragments.
// ws[0..383]  : Aprep[chunk][vgpr][lane]  (6 x 2 x 32 floats)
//               lane 0-15 -> M=lane (out ch), K = chunk*4 + v
//               lane16-31 -> M=lane-16,      K = chunk*4 + v + 2
// ws[384..399]: bias2[o] = scale2 * bias[o]
// ---------------------------------------------------------------------------
__global__ void prep_kernel(const float* __restrict__ w,
                            const float* __restrict__ bias,
                            const float* __restrict__ s1p,
                            const float* __restrict__ s2p,
                            float* __restrict__ ws) {
    int t = blockIdx.x * blockDim.x + threadIdx.x;
    float s1 = s1p[0], s2 = s2p[0];
    float s  = s1 * 0.125f * s2;
    if (t < 384) {
        int chunk = t >> 6;
        int v     = (t >> 5) & 1;
        int lane  = t & 31;
        int o     = lane & 15;
        int k     = chunk * 4 + v + ((lane >> 4) << 1);   // 0..23
        int c     = k >> 3;
        int rem   = k & 7;
        int dz = (rem >> 2) & 1, dy = (rem >> 1) & 1, dx = rem & 1;
        // effective tap: d==0 sums original taps {1,2}; d==1 takes tap {0}
        const int ksel[2][2] = { {1, 2}, {0, 0} };
        const int kcnt[2]    = { 2, 1 };
        float acc = 0.0f;
        for (int iz = 0; iz < kcnt[dz]; ++iz)
            for (int iy = 0; iy < kcnt[dy]; ++iy)
                for (int ix = 0; ix < kcnt[dx]; ++ix) {
                    int kz = ksel[dz][iz], ky = ksel[dy][iy], kx = ksel[dx][ix];
                    acc += w[(((c * COUT + o) * 3 + kz) * 3 + ky) * 3 + kx];
                }
        ws[t] = acc * s;
    } else if (t < 400) {
        int o = t - 384;
        ws[384 + o] = s2 * bias[o];
    }
}

// ---------------------------------------------------------------------------
// Main: one wave32 per output row (b, mz, my). GEMM M=16 (out ch), K=24,
// N=47 columns in 3 tiles of 16, via V_WMMA_F32_16X16X4_F32 (6 per tile).
// ---------------------------------------------------------------------------
__global__ __launch_bounds__(256) void conv_wmma_kernel(
        const float* __restrict__ x,
        const float* __restrict__ ws,
        float* __restrict__ out) {
    const int lane = threadIdx.x & 31;
    const int wave = threadIdx.x >> 5;
    const int row  = blockIdx.x * 8 + wave;
    if (row >= ROWS) return;                  // wave-uniform exit

    const int b  = row / (OUT_D * OUT_D);
    const int rr = row % (OUT_D * OUT_D);
    const int mz = rr / OUT_D;
    const int my = rr % OUT_D;

    const int n    = lane & 15;
    const int half = lane >> 4;

    // A fragments (scaled effective weights), per ISA 16x4 f32 A layout
    v2f a[NCHUNK];
#pragma unroll
    for (int ch = 0; ch < NCHUNK; ++ch) {
        a[ch].x = ws[ch * 64 +      lane];
        a[ch].y = ws[ch * 64 + 32 + lane];
    }

    // bias for each accumulator register's row M = r + 8*half
    float brow[8];
#pragma unroll
    for (int r = 0; r < 8; ++r)
        brow[r] = ws[384 + (half << 3) + r];

    // Per (chunk, vgpr): input row base + dx for this lane's K index
    int rowbase[NCHUNK][2];
    int dxs[NCHUNK][2];
#pragma unroll
    for (int ch = 0; ch < NCHUNK; ++ch) {
#pragma unroll
        for (int v = 0; v < 2; ++v) {
            int k   = ch * 4 + v + (half << 1);
            int c   = k >> 3;
            int rem = k & 7;
            int dz = (rem >> 2) & 1, dy = (rem >> 1) & 1;
            dxs[ch][v] = rem & 1;
            rowbase[ch][v] = ((b * CIN + c) * IN_D + (mz + dz)) * (IN_D * IN_D)
                           + (my + dy) * IN_D;
        }
    }

    const long outRowBase = (long)b * COUT * OUT_CH_STRIDE
                          + (long)mz * (OUT_D * OUT_D)
                          + (long)my * OUT_D;

    for (int mx0 = 0; mx0 < OUT_D; mx0 += 16) {
        v8f acc = {0.f, 0.f, 0.f, 0.f, 0.f, 0.f, 0.f, 0.f};
#pragma unroll
        for (int ch = 0; ch < NCHUNK; ++ch) {
            int xi0 = mx0 + n + dxs[ch][0]; if (xi0 > IN_D - 1) xi0 = IN_D - 1;
            int xi1 = mx0 + n + dxs[ch][1]; if (xi1 > IN_D - 1) xi1 = IN_D - 1;
            v2f bf;
            bf.x = x[rowbase[ch][0] + xi0];
            bf.y = x[rowbase[ch][1] + xi1];
            acc = __builtin_amdgcn_wmma_f32_16x16x4_f32(
                    /*neg_a=*/false, a[ch], /*neg_b=*/false, bf,
                    /*c_mod=*/(short)0, acc,
                    /*reuse_a=*/false, /*reuse_b=*/false);
        }
        if (mx0 + n < OUT_D) {
#pragma unroll
            for (int r = 0; r < 8; ++r) {
                int o = r + (half << 3);
                out[outRowBase + (long)o * OUT_CH_STRIDE + mx0 + n]
                    = acc[r] + brow[r];
            }
        }
    }
}

extern "C" void kernel_launch(void* const* d_in, const int* in_sizes, int n_in,
                              void* d_out, int out_size, void* d_ws, size_t ws_size,
                              hipStream_t stream) {
    (void)in_sizes; (void)n_in; (void)out_size; (void)ws_size;
    const float* x      = (const float*)d_in[0];
    const float* weight = (const float*)d_in[1];
    const float* bias   = (const float*)d_in[2];
    const float* s1     = (const float*)d_in[3];
    const float* s2     = (const float*)d_in[4];
    float*       out    = (float*)d_out;
    float*       ws     = (float*)d_ws;

    prep_kernel<<<1, 512, 0, stream>>>(weight, bias, s1, s2, ws);

    const int nblocks = (ROWS + 7) / 8;   // 8 waves of 32 per block
    conv_wmma_kernel<<<nblocks, 256, 0, stream>>>(x, ws, out);
}